// Dice_metric_56238301774607
// MI455X (gfx1250) — compile-verified
//
#include <hip/hip_runtime.h>
#include <hip/hip_bf16.h>

// Problem constants (B=16, C=8, H=W=512)
#define C_CLS 8
#define HW (512 * 512)
#define BLK_PER_IMG 64
#define NTHREADS 256
#define TILE 512
#define NTILES ((HW / BLK_PER_IMG) / TILE) // 8 tiles of 512 pixels per block

#if defined(__HIP_DEVICE_COMPILE__) && defined(__gfx1250__)
#define USE_ASYNC 1
#else
#define USE_ASYNC 0
#endif

// 16-byte global -> LDS copy (async on gfx1250: uses ASYNCcnt, no VGPR data)
__device__ __forceinline__ void cp16(const void* g, void* l) {
#if USE_ASYNC
  asm volatile("global_load_async_to_lds_b128 %0, %1, off"
               :
               : "v"((unsigned)(size_t)l), "v"((unsigned long long)(size_t)g)
               : "memory");
#else
  *(float4*)l = *(const float4*)g;
#endif
}

// 8-byte global -> LDS copy
__device__ __forceinline__ void cp8(const void* g, void* l) {
#if USE_ASYNC
  asm volatile("global_load_async_to_lds_b64 %0, %1, off"
               :
               : "v"((unsigned)(size_t)l), "v"((unsigned long long)(size_t)g)
               : "memory");
#else
  *(int2*)l = *(const int2*)g;
#endif
}

// Counters layout in workspace: cnt[b*24 + {0..7:pred, 8..15:tgt, 16..23:tps}]
__global__ __launch_bounds__(NTHREADS) void dice_count_kernel(
    const float* __restrict__ in, const int* __restrict__ tg,
    unsigned* __restrict__ cnt) {
  __shared__ __align__(16) float sF[2][C_CLS][TILE]; // 32 KB
  __shared__ __align__(16) int sT[2][TILE];          // 4 KB
  __shared__ unsigned blk[24];

  const int t = threadIdx.x;
  const int b = blockIdx.x >> 6;                 // image index
  const int ch = blockIdx.x & (BLK_PER_IMG - 1); // chunk within image
  const size_t inBase = (size_t)b * C_CLS * HW;
  const size_t tgBase = (size_t)b * HW;
  const int pix0 = ch * (HW / BLK_PER_IMG); // 4096 pixels per block

  if (t < 24) blk[t] = 0;

  unsigned cP[8] = {0, 0, 0, 0, 0, 0, 0, 0};
  unsigned cT[8] = {0, 0, 0, 0, 0, 0, 0, 0};
  unsigned cM[8] = {0, 0, 0, 0, 0, 0, 0, 0};
  unsigned accP = 0, accT = 0, accM = 0;

  // Each wave issues exactly 5 async ops per tile (4x b128 floats + 1x b64 tgts)
  auto issue = [&](int bufI, int tileStart) {
#pragma unroll
    for (int i = 0; i < 4; ++i) {
      const int k = t + NTHREADS * i; // 0..1023 chunk id
      const int c = k >> 7;           // channel 0..7
      const int j = k & 127;          // 16B chunk within channel row
      cp16(in + inBase + (size_t)c * HW + tileStart + j * 4,
           &sF[bufI][c][j * 4]);
    }
    cp8(tg + tgBase + tileStart + t * 2, &sT[bufI][t * 2]);
  };

  issue(0, pix0);
  int buf = 0;
  for (int ti = 0; ti < NTILES; ++ti) {
    if (ti + 1 < NTILES) {
      issue(buf ^ 1, pix0 + (ti + 1) * TILE);
#if USE_ASYNC
      // 5 newly issued remain in flight; previous tile's 5 have completed.
      asm volatile("s_wait_asynccnt 5" ::: "memory");
#endif
    } else {
#if USE_ASYNC
      asm volatile("s_wait_asynccnt 0" ::: "memory");
#endif
    }
    __syncthreads();

    // Two adjacent pixels per thread; ds_load_b64 per channel (conflict-free)
    float2 v[8];
#pragma unroll
    for (int c = 0; c < 8; ++c) v[c] = *(const float2*)&sF[buf][c][2 * t];
    const int tcA = sT[buf][2 * t];
    const int tcB = sT[buf][2 * t + 1];

    int pcA = 0;
    float bA = v[0].x;
    int pcB = 0;
    float bB = v[0].y;
#pragma unroll
    for (int c = 1; c < 8; ++c) { // strict '>' keeps first max (jnp.argmax)
      if (v[c].x > bA) { bA = v[c].x; pcA = c; }
      if (v[c].y > bB) { bB = v[c].y; pcB = c; }
    }

    // Packed nibble histograms: 4 bits per class, flushed every 8 pixels
    accP += (1u << (pcA * 4)) + (1u << (pcB * 4));
    accT += (1u << (tcA * 4)) + (1u << (tcB * 4));
    accM += (pcA == tcA ? (1u << (pcA * 4)) : 0u) +
            (pcB == tcB ? (1u << (pcB * 4)) : 0u);

    if (ti == 3 || ti == NTILES - 1) {
#pragma unroll
      for (int c = 0; c < 8; ++c) {
        cP[c] += (accP >> (c * 4)) & 0xFu;
        cT[c] += (accT >> (c * 4)) & 0xFu;
        cM[c] += (accM >> (c * 4)) & 0xFu;
      }
      accP = accT = accM = 0;
    }

    __syncthreads();
    buf ^= 1;
  }

  // Wave reduction (wave32), then block reduction in LDS, then global atomics
#pragma unroll
  for (int c = 0; c < 8; ++c) {
    unsigned p = cP[c], q = cT[c], m = cM[c];
#pragma unroll
    for (int off = 16; off > 0; off >>= 1) {
      p += __shfl_down(p, off, 32);
      q += __shfl_down(q, off, 32);
      m += __shfl_down(m, off, 32);
    }
    if ((t & 31) == 0) {
      atomicAdd(&blk[c], p);
      atomicAdd(&blk[8 + c], q);
      atomicAdd(&blk[16 + c], m);
    }
  }
  __syncthreads();
  if (t < 24) atomicAdd(&cnt[b * 24 + t], blk[t]);
}

__global__ void dice_zero_kernel(unsigned* cnt) {
  if (threadIdx.x < 16 * 24) cnt[threadIdx.x] = 0;
}

__global__ void dice_final_kernel(const unsigned* __restrict__ cnt,
                                  float* __restrict__ out) {
  const int b = threadIdx.x;
  if (b < 16) {
    const unsigned* c = cnt + b * 24;
    float acc = 0.f;
#pragma unroll
    for (int k = 1; k < 8; ++k) { // skip class 0, mean over classes 1..7
      const float tps = (float)c[16 + k];
      const float den = (float)(c[k] + c[8 + k]) + 1e-5f; // 2tps+fps+fns+eps
      acc += (2.f * tps) / den;
    }
    out[b] = acc * (1.f / 7.f);
  }
}

extern "C" void kernel_launch(void* const* d_in, const int* in_sizes, int n_in,
                              void* d_out, int out_size, void* d_ws,
                              size_t ws_size, hipStream_t stream) {
  (void)in_sizes;
  (void)n_in;
  (void)out_size;
  (void)ws_size;
  const float* in = (const float*)d_in[0];
  const int* tg = (const int*)d_in[1]; // integer input -> int32 per harness
  float* out = (float*)d_out;
  unsigned* cnt = (unsigned*)d_ws; // 16*24 uints = 1536 bytes of scratch

  dice_zero_kernel<<<1, 384, 0, stream>>>(cnt);
  dice_count_kernel<<<16 * BLK_PER_IMG, NTHREADS, 0, stream>>>(in, tg, cnt);
  dice_final_kernel<<<1, 32, 0, stream>>>(cnt, out);
}